// Attention_8589935148
// MI455X (gfx1250) — compile-verified
//
#include <hip/hip_runtime.h>
#include <hip/hip_bf16.h>
#include <math.h>

// Problem constants from the reference.
#define Bc 2
#define Hc 16
#define Sc 2048
#define Dc 64
#define QT (Sc / 16)   // 128 q-tiles
#define KT (Sc / 16)   // 128 k-tiles
#define NEGV (-1.0e9f)
#define SCALE 0.125f   // 1/sqrt(64)

typedef float v2f __attribute__((ext_vector_type(2)));
typedef float v8f __attribute__((ext_vector_type(8)));

// -----------------------------------------------------------------------------
// Kernel 1: scores = (Q K^T)*scale, mask, softmax over the HEAD axis, write P.
// Grid: B * QT * KT blocks; block = 512 threads = 16 waves (wave h = head h).
// Each wave computes one 16x16 f32 score tile via 16x V_WMMA_F32_16X16X4_F32.
// -----------------------------------------------------------------------------
__global__ __launch_bounds__(512)
void attn_scores_headsoftmax(const float* __restrict__ Q,
                             const float* __restrict__ K,
                             const int*   __restrict__ mask,
                             float*       __restrict__ P) {
  const int blk = blockIdx.x;
  const int kt  = blk % KT;
  const int qt  = (blk / KT) % QT;
  const int b   = blk / (KT * QT);

  const int tid  = threadIdx.x;
  const int h    = tid >> 5;        // wave index == head
  const int lane = tid & 31;
  const int half = lane >> 4;       // 0: lanes 0-15, 1: lanes 16-31
  const int l16  = lane & 15;

  // Padded to dodge LDS bank conflicts between lane halves.
  __shared__ float lds_s[Hc][16][17];

  // --- Phase 1: per-head 16x16 score tile via f32 WMMA ---------------------
  const float* Qb = Q + (((size_t)b * Hc + h) * Sc + (size_t)qt * 16) * Dc;
  const float* Kb = K + (((size_t)b * Hc + h) * Sc + (size_t)kt * 16) * Dc;

  // f32 A-matrix (16x4) layout: lane(0-15)=row M, VGPR0/1 = K 0,1 (low half)
  // or K 2,3 (high half)  ->  per lane: 2 contiguous floats at col 4*kk+2*half.
  // B = K^T has the identical per-lane load pattern over K's rows.
  const int col0 = 2 * half;

  v8f c = {};
#pragma unroll
  for (int kk = 0; kk < 16; ++kk) {
    v2f a  = *(const v2f*)(Qb + (size_t)l16 * Dc + kk * 4 + col0);
    v2f bm = *(const v2f*)(Kb + (size_t)l16 * Dc + kk * 4 + col0);
    c = __builtin_amdgcn_wmma_f32_16x16x4_f32(false, a, false, bm,
                                              (short)0, c, false, false);
  }

  // Scale + mask, stash tile in LDS.  C/D layout: VGPR r, low half -> M=r,
  // high half -> M=r+8, N = lane%16.
  const int* mrow = mask + ((size_t)b * Sc + (size_t)qt * 16) * Sc
                         + (size_t)kt * 16;
#pragma unroll
  for (int r = 0; r < 8; ++r) {
    const int m  = r + 8 * half;
    float s      = c[r] * SCALE;
    const int mv = mrow[(size_t)m * Sc + l16];
    lds_s[h][m][l16] = (mv == 0) ? NEGV : s;
  }
  __syncthreads();

  // --- Phase 2: softmax across the 16 heads at each (q,k) ------------------
  if (tid < 256) {
    const int M = tid >> 4;
    const int N = tid & 15;

    float v[Hc];
    float mx = -3.4e38f;
#pragma unroll
    for (int hh = 0; hh < Hc; ++hh) {
      v[hh] = lds_s[hh][M][N];
      mx    = fmaxf(mx, v[hh]);
    }
    float sum = 0.0f;
#pragma unroll
    for (int hh = 0; hh < Hc; ++hh) {
      const float e = __expf(v[hh] - mx);
      v[hh] = e;
      sum  += e;
    }
    const float inv = 1.0f / sum;

    const size_t base0 = ((size_t)b * Hc) * Sc * Sc
                       + (size_t)(qt * 16 + M) * Sc + (size_t)kt * 16 + N;
#pragma unroll
    for (int hh = 0; hh < Hc; ++hh) {
      P[base0 + (size_t)hh * Sc * Sc] = v[hh] * inv;
    }
  }
}

// -----------------------------------------------------------------------------
// Kernel 2: O = P @ V per (b,h).  Grid: B*H*QT blocks; block = 128 threads =
// 4 waves; wave w owns output d-tile w (16 columns of D=64).
// k-loop accumulates 16x16 output with V_WMMA_F32_16X16X4_F32.
// -----------------------------------------------------------------------------
__global__ __launch_bounds__(128)
void attn_pv(const float* __restrict__ P,
             const float* __restrict__ V,
             float*       __restrict__ O) {
  const int blk = blockIdx.x;
  const int qt  = blk % QT;
  const int h   = (blk / QT) % Hc;
  const int b   = blk / (QT * Hc);

  const int tid  = threadIdx.x;
  const int w    = tid >> 5;        // d-tile index 0..3
  const int lane = tid & 31;
  const int half = lane >> 4;
  const int l16  = lane & 15;
  const int d0   = w * 16;

  const float* Pb = P + (((size_t)b * Hc + h) * Sc + (size_t)qt * 16) * Sc;
  const float* Vb = V + (((size_t)b * Hc + h) * Sc) * Dc;

  v8f c = {};
  for (int k = 0; k < Sc; k += 16) {
    // Stream-prefetch the P rows ~1KB ahead (global_prefetch_b8).
    __builtin_prefetch(Pb + (size_t)l16 * Sc + k + 256, 0, 1);
#pragma unroll
    for (int kk = 0; kk < 16; kk += 4) {
      const int kb = k + kk;
      // A = P tile rows (contiguous b64 per lane).
      v2f a = *(const v2f*)(Pb + (size_t)l16 * Sc + kb + 2 * half);
      // B = V[kb..kb+3][d0..d0+15]; per lane: rows 2*half, 2*half+1, col l16.
      v2f bm;
      bm.x = Vb[(size_t)(kb + 2 * half)     * Dc + d0 + l16];
      bm.y = Vb[(size_t)(kb + 2 * half + 1) * Dc + d0 + l16];
      c = __builtin_amdgcn_wmma_f32_16x16x4_f32(false, a, false, bm,
                                                (short)0, c, false, false);
    }
  }

  float* Ob = O + (((size_t)b * Hc + h) * Sc + (size_t)qt * 16) * Dc + d0;
#pragma unroll
  for (int r = 0; r < 8; ++r) {
    Ob[(size_t)(r + 8 * half) * Dc + l16] = c[r];
  }
}

// -----------------------------------------------------------------------------
extern "C" void kernel_launch(void* const* d_in, const int* in_sizes, int n_in,
                              void* d_out, int out_size, void* d_ws, size_t ws_size,
                              hipStream_t stream) {
  (void)in_sizes; (void)n_in; (void)out_size; (void)d_ws; (void)ws_size;

  const float* Q    = (const float*)d_in[0];
  const float* K    = (const float*)d_in[1];
  const float* V    = (const float*)d_in[2];
  const int*   mask = (const int*)d_in[3];

  // d_out = [ O : B*H*S*D floats | P : B*H*S*S floats ]
  float* O = (float*)d_out;
  float* P = O + (size_t)Bc * Hc * Sc * Dc;

  // Kernel 1: fused scores + mask + head-axis softmax -> P (no scratch needed).
  attn_scores_headsoftmax<<<Bc * QT * KT, 512, 0, stream>>>(Q, K, mask, P);

  // Kernel 2: O = P @ V.
  attn_pv<<<Bc * Hc * QT, 128, 0, stream>>>(P, V, O);
}